// Net_27865747816550
// MI455X (gfx1250) — compile-verified
//
#include <hip/hip_runtime.h>
#include <math.h>

// Problem constants (match reference).
#define NN      100000
#define FIN     256
#define HIDF    16
#define NCLSF   10
#define KST     3
#define LDC     48          // 3 stacks x 16 padded feature columns

typedef __attribute__((ext_vector_type(2))) float v2f;
typedef __attribute__((ext_vector_type(8))) float v8f;

// ---------------------------------------------------------------------------
// Normalization: deg -> dis -> norm
// ---------------------------------------------------------------------------
__global__ void __launch_bounds__(256)
deg_kernel(const int* __restrict__ col, float* __restrict__ deg, int E) {
    int e = blockIdx.x * blockDim.x + threadIdx.x;
    if (e < E) unsafeAtomicAdd(&deg[col[e]], 1.0f);
}

__global__ void __launch_bounds__(256)
dis_kernel(const float* __restrict__ deg, float* __restrict__ dis, int n) {
    int i = blockIdx.x * blockDim.x + threadIdx.x;
    if (i < n) { float d = deg[i]; dis[i] = (d > 0.f) ? rsqrtf(d) : 0.f; }
}

__global__ void __launch_bounds__(256)
norm_kernel(const int* __restrict__ row, const int* __restrict__ col,
            const float* __restrict__ dis, float* __restrict__ nrm, int E) {
    int e = blockIdx.x * blockDim.x + threadIdx.x;
    if (e < E) nrm[e] = dis[row[e]] * dis[col[e]];
}

// ---------------------------------------------------------------------------
// fp32 WMMA GEMM (V_WMMA_F32_16X16X4_F32), compile-time reduction depth KD.
//
// Block = 128 threads = 4 waves. Each wave computes one 16(M) x 16(N) tile of
// C[m, kstk*16 + n] = sum_k A[m, a_off + k] * B_kstk[k, n] (+bias).
// grid = (ceil(nrows/64), KST).
//
// All tail/padding handling happens at LDS staging time (zero-fill), so the
// inner loop is branch-free: ds_load_b64 (A frag) + ds_load_b64 (B^T frag)
// + v_wmma, fully unrolled. EXEC is all-ones throughout the WMMA loop.
//
// Fragment layout per CDNA5 ISA 7.12.2 (32-bit ops):
//   A 16x4 : lane L -> M = L&15, VGPR pair holds K = {0,1} (L<16) / {2,3}
//   B 4x16 : lane L -> N = L&15, same K split  (stored transposed in LDS)
//   C/D    : lane L -> N = L&15, VGPR j holds M = j + 8*(L>=16)
// Row stride KD+4 keeps LDS banks conflict-free and 8-byte alignment (even).
// ---------------------------------------------------------------------------
template<int KD>
__global__ void __launch_bounds__(128)
wmma_gemm_f32(const float* __restrict__ A, int lda, int a_off_stride,
              const float* __restrict__ B, int b_stack_stride, int ldb,
              int k_valid, int n_valid,
              const float* __restrict__ bias, int bias_stride,
              float* __restrict__ C, int ldc, int nrows) {
    constexpr int SA = KD + 4;
    __shared__ __align__(16) float sBt[16 * SA];      // B transposed, zero-padded
    __shared__ __align__(16) float sA[4][16 * SA];    // per-wave A tiles

    const int tid   = threadIdx.x;
    const int wave  = tid >> 5;
    const int lane  = tid & 31;
    const int kstk  = blockIdx.y;
    const int m0    = (blockIdx.x * 4 + wave) * 16;
    const int a_off = kstk * a_off_stride;
    const float* __restrict__ Bk = B + (size_t)kstk * b_stack_stride;

    // Stage B^T cooperatively (all 128 threads), zero-padding invalid k/n.
    for (int idx = tid; idx < 16 * KD; idx += 128) {
        int n  = idx / KD;
        int kk = idx - n * KD;
        sBt[n * SA + kk] = (n < n_valid && kk < k_valid)
                         ? Bk[(size_t)kk * ldb + n] : 0.f;
    }
    // Stage this wave's 16 x KD A tile (coalesced within the wave).
    for (int idx = lane; idx < 16 * KD; idx += 32) {
        int r  = idx / KD;
        int cc = idx - r * KD;
        int gr = m0 + r;
        sA[wave][r * SA + cc] = (gr < nrows) ? A[(size_t)gr * lda + a_off + cc] : 0.f;
    }
    __syncthreads();

    const int half = lane >> 4;
    const int lrow = lane & 15;
    const int koff = half ? 2 : 0;
    const float* a_base = &sA[wave][lrow * SA + koff];
    const float* b_base = &sBt[lrow * SA + koff];

    v8f acc = {0.f, 0.f, 0.f, 0.f, 0.f, 0.f, 0.f, 0.f};
    #pragma unroll
    for (int kk = 0; kk < KD; kk += 4) {
        v2f a = *(const v2f*)(a_base + kk);       // ds_load_b64
        v2f b = *(const v2f*)(b_base + kk);       // ds_load_b64
        acc = __builtin_amdgcn_wmma_f32_16x16x4_f32(
                  false, a, false, b, (short)0, acc, false, false);
    }

    if (lrow < n_valid) {
        float bv = bias ? bias[kstk * bias_stride + lrow] : 0.f;
        #pragma unroll
        for (int j = 0; j < 8; ++j) {
            int m = m0 + j + (half ? 8 : 0);
            if (m < nrows) C[(size_t)m * ldc + kstk * 16 + lrow] = acc[j] + bv;
        }
    }
}

// ---------------------------------------------------------------------------
// Edge propagation (scatter-add). Layer 1: all 48 cols valid, float4 chunks.
// ---------------------------------------------------------------------------
__global__ void __launch_bounds__(256)
scatter_l1(const float* __restrict__ h, const int* __restrict__ row,
           const int* __restrict__ col, const float* __restrict__ nrm,
           float* __restrict__ agg, int E) {
    long long t = (long long)blockIdx.x * blockDim.x + threadIdx.x;
    if (t >= (long long)E * 12) return;
    int e = (int)(t / 12);
    int j = (int)(t - (long long)e * 12);
    float w = nrm[e];
    if (w == 0.f) return;
    int r = row[e], c = col[e];
    const float4 v = *(const float4*)(h + (size_t)r * LDC + j * 4);
    float* d = agg + (size_t)c * LDC + j * 4;
    unsafeAtomicAdd(d + 0, v.x * w);
    unsafeAtomicAdd(d + 1, v.y * w);
    unsafeAtomicAdd(d + 2, v.z * w);
    unsafeAtomicAdd(d + 3, v.w * w);
}

// Layer 2: only cols (k*16 + c), c < 10, are valid.
__global__ void __launch_bounds__(256)
scatter_l2(const float* __restrict__ h, const int* __restrict__ row,
           const int* __restrict__ col, const float* __restrict__ nrm,
           float* __restrict__ agg, int E) {
    long long t = (long long)blockIdx.x * blockDim.x + threadIdx.x;
    if (t >= (long long)E * 30) return;
    int e = (int)(t / 30);
    int j = (int)(t - (long long)e * 30);
    float w = nrm[e];
    if (w == 0.f) return;
    int k = j / 10, c = j - k * 10;
    int r = row[e], cl = col[e];
    float v = h[(size_t)r * LDC + k * 16 + c] * w;
    unsafeAtomicAdd(&agg[(size_t)cl * LDC + k * 16 + c], v);
}

// ---------------------------------------------------------------------------
// Elementwise stages
// ---------------------------------------------------------------------------
__global__ void __launch_bounds__(256)
relu_add_kernel(const float* __restrict__ agg, const float* __restrict__ root,
                float* __restrict__ out, int n) {
    int i = blockIdx.x * blockDim.x + threadIdx.x;
    if (i < n) { float v = agg[i] + root[i]; out[i] = v > 0.f ? v : 0.f; }
}

// h[n,c] = mean_k relu(agg[n,k*16+c] + root[n,k*16+c])  (outer relu is a no-op)
__global__ void __launch_bounds__(256)
mean_relu_kernel(const float* __restrict__ agg, const float* __restrict__ root,
                 float* __restrict__ h, int n) {
    int t = blockIdx.x * blockDim.x + threadIdx.x;
    if (t >= n * HIDF) return;
    int nn = t >> 4, c = t & 15;
    float s = 0.f;
    #pragma unroll
    for (int k = 0; k < KST; ++k) {
        size_t idx = (size_t)nn * LDC + k * 16 + c;
        float v = agg[idx] + root[idx];
        s += (v > 0.f) ? v : 0.f;
    }
    h[t] = s * (1.f / 3.f);
}

// out2 = agg + root on valid cols (c%16 < 10), zero elsewhere (keeps next GEMM's A clean)
__global__ void __launch_bounds__(256)
add_mask_kernel(const float* __restrict__ agg, const float* __restrict__ root,
                float* __restrict__ out, int n) {
    int i = blockIdx.x * blockDim.x + threadIdx.x;
    if (i >= n) return;
    int cc = (i % LDC) & 15;
    out[i] = (cc < NCLSF) ? (agg[i] + root[i]) : 0.f;
}

// Final: mean over K then log_softmax over 10 classes.
__global__ void __launch_bounds__(256)
finalize_kernel(const float* __restrict__ agg, const float* __restrict__ root,
                float* __restrict__ out, int n) {
    int i = blockIdx.x * blockDim.x + threadIdx.x;
    if (i >= n) return;
    float z[NCLSF];
    #pragma unroll
    for (int c = 0; c < NCLSF; ++c) {
        float s = 0.f;
        #pragma unroll
        for (int k = 0; k < KST; ++k) {
            size_t idx = (size_t)i * LDC + k * 16 + c;
            s += agg[idx] + root[idx];
        }
        z[c] = s * (1.f / 3.f);
    }
    float mx = z[0];
    #pragma unroll
    for (int c = 1; c < NCLSF; ++c) mx = fmaxf(mx, z[c]);
    float se = 0.f;
    #pragma unroll
    for (int c = 0; c < NCLSF; ++c) se += expf(z[c] - mx);
    float l = mx + logf(se);
    #pragma unroll
    for (int c = 0; c < NCLSF; ++c) out[(size_t)i * NCLSF + c] = z[c] - l;
}

// ---------------------------------------------------------------------------
// Host orchestration
// ---------------------------------------------------------------------------
static inline int cdiv(long long a, long long b) { return (int)((a + b - 1) / b); }

extern "C" void kernel_launch(void* const* d_in, const int* in_sizes, int n_in,
                              void* d_out, int out_size, void* d_ws, size_t ws_size,
                              hipStream_t stream) {
    const float* x   = (const float*)d_in[0];
    const int*   ei  = (const int*)d_in[1];
    const float* w1i = (const float*)d_in[2];   // [3,256,16]
    const float* w1  = (const float*)d_in[3];   // [1,3,16,16]
    const float* w1r = (const float*)d_in[4];   // [1,3,256,16]
    const float* b1  = (const float*)d_in[5];   // [1,3,1,16]
    const float* w2i = (const float*)d_in[6];   // [3,16,10]
    const float* w2  = (const float*)d_in[7];   // [1,3,10,10]
    const float* w2r = (const float*)d_in[8];   // [1,3,16,10]
    const float* b2  = (const float*)d_in[9];   // [1,3,1,10]
    float* out = (float*)d_out;

    const int E = in_sizes[1] / 2;
    const int* row = ei;
    const int* col = ei + E;

    // Workspace carve-out (floats): norm[E] | bufA | bufB | bufC | bufD [N*48 each] | h [N*16]
    float* ws   = (float*)d_ws;
    float* nrm  = ws;
    float* bufA = nrm  + (size_t)E;              // h0 / h1 / h0_2 / h1_2
    float* bufB = bufA + (size_t)NN * LDC;       // root1, then root2
    float* bufC = bufB + (size_t)NN * LDC;       // agg (re-zeroed per propagation)
    float* bufD = bufC + (size_t)NN * LDC;       // out1, then out2
    float* hbuf = bufD + (size_t)NN * LDC;       // layer-1 output [N,16]
    float* deg  = bufA;                          // transient aliases (dead before GEMMs)
    float* dis  = bufA + NN;

    // ---- edge norm ----
    hipMemsetAsync(deg, 0, (size_t)NN * sizeof(float), stream);
    deg_kernel <<<cdiv(E, 256),  256, 0, stream>>>(col, deg, E);
    dis_kernel <<<cdiv(NN, 256), 256, 0, stream>>>(deg, dis, NN);
    norm_kernel<<<cdiv(E, 256),  256, 0, stream>>>(row, col, dis, nrm, E);

    dim3 ggrid(cdiv(NN, 64), KST);
    const size_t aggBytes = (size_t)NN * LDC * sizeof(float);

    // ---- layer 1 ----
    wmma_gemm_f32<FIN><<<ggrid, 128, 0, stream>>>(x, FIN, 0, w1i, FIN * HIDF, HIDF,
                                                  FIN, HIDF, nullptr, 0,
                                                  bufA, LDC, NN);                 // h0
    wmma_gemm_f32<FIN><<<ggrid, 128, 0, stream>>>(x, FIN, 0, w1r, FIN * HIDF, HIDF,
                                                  FIN, HIDF, b1, HIDF,
                                                  bufB, LDC, NN);                 // root1
    hipMemsetAsync(bufC, 0, aggBytes, stream);
    scatter_l1<<<cdiv((long long)E * 12, 256), 256, 0, stream>>>(bufA, row, col, nrm, bufC, E);
    relu_add_kernel<<<cdiv((long long)NN * LDC, 256), 256, 0, stream>>>(bufC, bufB, bufD, NN * LDC);
    wmma_gemm_f32<16><<<ggrid, 128, 0, stream>>>(bufD, LDC, 16, w1, HIDF * HIDF, HIDF,
                                                 HIDF, HIDF, nullptr, 0,
                                                 bufA, LDC, NN);                  // h1
    hipMemsetAsync(bufC, 0, aggBytes, stream);
    scatter_l1<<<cdiv((long long)E * 12, 256), 256, 0, stream>>>(bufA, row, col, nrm, bufC, E);
    mean_relu_kernel<<<cdiv((long long)NN * HIDF, 256), 256, 0, stream>>>(bufC, bufB, hbuf, NN);

    // ---- layer 2 ----
    wmma_gemm_f32<16><<<ggrid, 128, 0, stream>>>(hbuf, HIDF, 0, w2i, HIDF * NCLSF, NCLSF,
                                                 HIDF, NCLSF, nullptr, 0,
                                                 bufA, LDC, NN);                  // h0_2
    wmma_gemm_f32<16><<<ggrid, 128, 0, stream>>>(hbuf, HIDF, 0, w2r, HIDF * NCLSF, NCLSF,
                                                 HIDF, NCLSF, b2, NCLSF,
                                                 bufB, LDC, NN);                  // root2
    hipMemsetAsync(bufC, 0, aggBytes, stream);
    scatter_l2<<<cdiv((long long)E * 30, 256), 256, 0, stream>>>(bufA, row, col, nrm, bufC, E);
    add_mask_kernel<<<cdiv((long long)NN * LDC, 256), 256, 0, stream>>>(bufC, bufB, bufD, NN * LDC);
    wmma_gemm_f32<16><<<ggrid, 128, 0, stream>>>(bufD, LDC, 16, w2, NCLSF * NCLSF, NCLSF,
                                                 NCLSF, NCLSF, nullptr, 0,
                                                 bufA, LDC, NN);                  // h1_2
    hipMemsetAsync(bufC, 0, aggBytes, stream);
    scatter_l2<<<cdiv((long long)E * 30, 256), 256, 0, stream>>>(bufA, row, col, nrm, bufC, E);
    finalize_kernel<<<cdiv(NN, 256), 256, 0, stream>>>(bufC, bufB, out, NN);
}